// QCONV2d_44598940401781
// MI455X (gfx1250) — compile-verified
//
#include <hip/hip_runtime.h>

typedef __attribute__((ext_vector_type(2))) float v2f;
typedef __attribute__((ext_vector_type(4))) float v4f;
typedef __attribute__((ext_vector_type(8))) float v8f;

#define NQ      9
#define C_IN    8
#define HW      32
#define NFILT   16
#define INV2PI  0.15915494309189535f  // v_cos_f32 takes input in turns

// z(b,c,i,j) = prod_{q=1..8} cos(x_pad[b,c,i+dy,j+dx] + w[b,q])
// (qubit 0 / top-left patch element cancels out of the conjugated observable)
__device__ __forceinline__ float zprod(const float* __restrict__ x,
                                       const float wq[NQ],
                                       int b, int c, int i, int j) {
  const float* xc = x + ((size_t)(b * C_IN + c)) * HW * HW;
  float prod = 1.0f;
#pragma unroll
  for (int a = 0; a < 3; ++a) {
#pragma unroll
    for (int bb = 0; bb < 3; ++bb) {
      int q = a * 3 + bb;
      if (q == 0) continue;               // Z_0 drops out
      int ii = i + a - 1;
      int jj = j + bb - 1;
      bool inb = (ii >= 0) & (ii < HW) & (jj >= 0) & (jj < HW);
      float xv = inb ? xc[ii * HW + jj] : 0.0f;   // zero padding
      prod *= __builtin_amdgcn_cosf((xv + wq[q]) * INV2PI);
    }
  }
  return prod;
}

// One wave32 handles 16 consecutive pixels of one (batch,row).
// Channel sum + 16-filter broadcast done with V_WMMA_F32_16X16X4_F32 x2
// against an all-ones B matrix: D[m][n] = sum_c z[pixel m][c] for every n,
// and column n is written out as filter n (fully coalesced b128 stores).
__global__ void __launch_bounds__(128)
qconv_vqc_kernel(const float* __restrict__ x,
                 const float* __restrict__ w,
                 float* __restrict__ out) {
  int flat = blockIdx.x * blockDim.x + threadIdx.x;
  int lane = flat & 31;
  int wv   = flat >> 5;            // 0..255 waves total
  int m    = lane & 15;            // A-matrix row = pixel-in-wave
  int half = lane >> 4;            // 0: K=0,1   1: K=2,3
  int b     = wv >> 6;             // 4 batches x 64 waves
  int pbase = (wv & 63) * 16;      // pixel base inside 32x32 image
  int p = pbase + m;
  int i = p >> 5;
  int j = p & 31;

  float wq[NQ];
#pragma unroll
  for (int q = 1; q < NQ; ++q) wq[q] = w[b * NQ + q];

  // A 16x4 f32 layout: lanes 0-15 -> (K=0,K=1); lanes 16-31 -> (K=2,K=3)
  v2f a0, a1, ones;
  a0.x = zprod(x, wq, b, 2 * half + 0, i, j);      // channels 0..3 (wmma #1)
  a0.y = zprod(x, wq, b, 2 * half + 1, i, j);
  a1.x = zprod(x, wq, b, 4 + 2 * half, i, j);      // channels 4..7 (wmma #2)
  a1.y = zprod(x, wq, b, 5 + 2 * half, i, j);
  ones.x = 1.0f; ones.y = 1.0f;

  v8f acc = {};
  acc = __builtin_amdgcn_wmma_f32_16x16x4_f32(false, a0, false, ones,
                                              (short)0, acc, false, false);
  acc = __builtin_amdgcn_wmma_f32_16x16x4_f32(false, a1, false, ones,
                                              (short)0, acc, false, false);

  // D layout: lanes 0-15: VGPR r = (M=r,   N=lane)    -> pixels pbase+0..7
  //           lanes16-31: VGPR r = (M=8+r, N=lane-16) -> pixels pbase+8..15
  // column N == output filter index; all columns identical by construction.
  int f = lane & 15;
  float* o = out + ((size_t)(b * NFILT + f)) * (HW * HW) + pbase + half * 8;
  v4f lo = {acc[0], acc[1], acc[2], acc[3]};
  v4f hi = {acc[4], acc[5], acc[6], acc[7]};
  *(v4f*)(o + 0) = lo;   // 32B-aligned (offset is a multiple of 8 floats)
  *(v4f*)(o + 4) = hi;
}

extern "C" void kernel_launch(void* const* d_in, const int* in_sizes, int n_in,
                              void* d_out, int out_size, void* d_ws, size_t ws_size,
                              hipStream_t stream) {
  (void)in_sizes; (void)n_in; (void)out_size; (void)d_ws; (void)ws_size;
  const float* x = (const float*)d_in[0];   // (4,8,32,32) f32
  const float* w = (const float*)d_in[1];   // (4,1,9)     f32
  float* out = (float*)d_out;               // (4,16,32,32) f32
  // 4096 pixels, 16 per wave -> 256 waves -> 8192 threads
  qconv_vqc_kernel<<<64, 128, 0, stream>>>(x, w, out);
}